// VanillaRNN_18511309046056
// MI455X (gfx1250) — compile-verified
//
#include <hip/hip_runtime.h>
#include <hip/hip_bf16.h>
#include <math.h>

typedef __attribute__((ext_vector_type(2))) float v2f;
typedef __attribute__((ext_vector_type(8))) float v8f;

namespace {
constexpr int T   = 256;    // SEQ_LEN
constexpr int H   = 1024;   // NUM_HIDDEN
constexpr int NB  = 1024;   // BATCH (== H)
constexpr int NC  = 10;     // NUM_CLASSES
constexpr int HPL = H / 32; // hidden elements per lane (wave32)
}

__device__ __forceinline__ float dtanh(float v) {
#if __has_builtin(__builtin_amdgcn_tanhf)
  return __builtin_amdgcn_tanhf(v);   // gfx1250 v_tanh_f32 (TRANS)
#else
  return tanhf(v);
#endif
}

// ---------------------------------------------------------------------------
// Phase 1: sequential recurrence. One wave32 per batch element b.
// Exploits rank-1 structure: ht[b,h] = tanh(a*Whx[h] + c). Per step we only
// need s = sum_h tanh(a*Whx[h]+c)*Whh[h]; weights live in VGPRs, the cross-
// lane reduction is a 5-stage wave32 butterfly. Emits only (a,c) per b.
// ---------------------------------------------------------------------------
__global__ __launch_bounds__(32) void rnn_recurrence(
    const float* __restrict__ x,    // [NB, T]
    const float* __restrict__ Whx,  // [1, H]
    const float* __restrict__ Whh,  // [1, H]
    const float* __restrict__ bh,   // [1, H]  (effective add is bh[0,b], B==H)
    float* __restrict__ ac)         // ws: a in [0,NB), c in [NB,2NB)
{
  const int b    = blockIdx.x;
  const int lane = threadIdx.x;     // 0..31

  float wx[HPL], wh[HPL];
#pragma unroll
  for (int j = 0; j < HPL; ++j) {
    wx[j] = Whx[j * 32 + lane];     // coalesced, kept in VGPRs
    wh[j] = Whh[j * 32 + lane];
  }
  const float bhb = bh[b];
  const float* xb = x + b * T;

  float a = xb[0];                  // h1 = tanh(a*Whx + c)
  float c = bhb;
  for (int t = 1; t < T; ++t) {
    float p = 0.0f;
#pragma unroll
    for (int j = 0; j < HPL; ++j)
      p += dtanh(fmaf(a, wx[j], c)) * wh[j];
    // wave32 butterfly reduction -> every lane holds the full sum
#pragma unroll
    for (int m = 16; m >= 1; m >>= 1)
      p += __shfl_xor(p, m, 32);
    c = bhb + p;
    a = xb[t];
  }
  if (lane == 0) {
    ac[b]      = a;                 // hT[b,h] = tanh(a*Whx[h] + c)
    ac[NB + b] = c;
  }
}

// ---------------------------------------------------------------------------
// Phase 2: pt[h,c] = sum_b hT[b,h]*Wph[b,c] + bp[c], via V_WMMA_F32_16X16X4_F32
// (pure f32 -> same precision class as the reference). One wave per 16-row
// h-tile; N tile is 16 wide, columns 0..9 valid. A-fragment values are
// regenerated on the fly from (a,c) staged in LDS -- no 4MB hT buffer.
//
// ISA fragment layouts (CDNA5 ISA):
//   A 16x4 f32 : element r -> A[m = lane&15, k = r + 2*(lane>>4)]
//   B  4x16 f32: element r -> B[k = r + 2*(lane>>4), n = lane&15]
//   C/D 16x16  : element r -> D[m = r + 8*(lane>>4), n = lane&15]
// ---------------------------------------------------------------------------
__global__ __launch_bounds__(32) void rnn_readout(
    const float* __restrict__ ac,   // ws: a[0..NB), c[NB..2NB)
    const float* __restrict__ Whx,  // [1, H]
    const float* __restrict__ Wph,  // [NB(=H), NC]
    const float* __restrict__ bp,   // [1, NC]
    float* __restrict__ out)        // [H, NC] row-major
{
  __shared__ float sa[NB];
  __shared__ float sc[NB];
  const int lane = threadIdx.x;
  for (int i = lane; i < NB; i += 32) {
    sa[i] = ac[i];
    sc[i] = ac[NB + i];
  }
  __syncthreads();

  const int h0 = blockIdx.x * 16;
  const int mn = lane & 15;               // m for A / n for B,D
  const int kh = (lane >> 4) * 2;         // k sub-offset within group of 4
  const float wxm   = Whx[h0 + mn];       // Whx value for this lane's A-row
  const bool  nval  = (mn < NC);
  const int   cidx  = nval ? mn : 0;      // branchless, always-valid load
  const float nmask = nval ? 1.0f : 0.0f;

  v8f acc = {};
  for (int k0 = 0; k0 < NB; k0 += 4) {
    const int k = k0 + kh;
    v2f afrag, bfrag;
    afrag.x = dtanh(fmaf(sa[k],     wxm, sc[k]));      // hT[k,   h0+mn]
    afrag.y = dtanh(fmaf(sa[k + 1], wxm, sc[k + 1]));  // hT[k+1, h0+mn]
    bfrag.x = Wph[k * NC + cidx] * nmask;              // Wph[k,   n]
    bfrag.y = Wph[(k + 1) * NC + cidx] * nmask;        // Wph[k+1, n]
    // 8 args: (neg_a, A, neg_b, B, c_mod, C, reuse_a, reuse_b)
    acc = __builtin_amdgcn_wmma_f32_16x16x4_f32(false, afrag, false, bfrag,
                                                (short)0, acc, false, false);
  }

  if (nval) {
    const float bpn = bp[mn];
#pragma unroll
    for (int r = 0; r < 8; ++r) {
      const int h = h0 + r + 8 * (lane >> 4);
      out[h * NC + mn] = acc[r] + bpn;
    }
  }
}

// ---------------------------------------------------------------------------
extern "C" void kernel_launch(void* const* d_in, const int* in_sizes, int n_in,
                              void* d_out, int out_size, void* d_ws, size_t ws_size,
                              hipStream_t stream) {
  (void)in_sizes; (void)n_in; (void)out_size; (void)ws_size;
  const float* x   = (const float*)d_in[0];
  const float* Whx = (const float*)d_in[1];
  const float* Whh = (const float*)d_in[2];
  const float* Wph = (const float*)d_in[3];
  const float* bh  = (const float*)d_in[4];
  const float* bp  = (const float*)d_in[5];
  float* out = (float*)d_out;
  float* ac  = (float*)d_ws;            // needs only 2*NB*4 = 8 KB scratch

  rnn_recurrence<<<dim3(NB),     dim3(32), 0, stream>>>(x, Whx, Whh, bh, ac);
  rnn_readout  <<<dim3(H / 16),  dim3(32), 0, stream>>>(ac, Whx, Wph, bp, out);
}